// DVS_89180700934556
// MI455X (gfx1250) — compile-verified
//
#include <hip/hip_runtime.h>
#include <math.h>

// ---------------- problem constants (match reference) ----------------
#define H_IMG   512
#define W_IMG   640
#define BORDER  5
#define CH      (H_IMG - 2*BORDER)      // 502
#define CW      (W_IMG - 2*BORDER)      // 630
#define NPIX    (CH * CW)               // 316260
#define BATCH   64
#define PXc     600.0f
#define PYc     600.0f
#define U0c     320.0f
#define V0c     256.0f

typedef __attribute__((ext_vector_type(2))) float v2f;
typedef __attribute__((ext_vector_type(8))) float v8f;

// Gram-accumulation kernel.
// grid = (gpb, BATCH), block = 256 (8 waves of 32).
// Each wave: 32 pixels/iter -> LDS transpose -> 8x V_WMMA_F32_16X16X4_F32
// accumulating the 7x7 augmented Gram [L | e]^T [L | e] in an 8-VGPR f32 C.
__global__ __launch_bounds__(256)
void dvs_gram_kernel(const float* __restrict__ Is,
                     const float* __restrict__ Ids,
                     const float* __restrict__ Zinv,
                     float* __restrict__ gpart, int gpb)
{
    // Per-wave transpose slab: 32 pixel slots x 17 floats (pad for banks).
    __shared__ float xpose[8][32][17];
    __shared__ float part[8][49];

    const int b    = blockIdx.y;
    const int gx   = blockIdx.x;
    const int tid  = threadIdx.x;
    const int wave = tid >> 5;
    const int lane = tid & 31;

    const float* pIs  = Is   + (size_t)b * H_IMG * W_IMG;
    const float* pIds = Ids  + (size_t)b * H_IMG * W_IMG;
    const float* pZ   = Zinv + (size_t)b * H_IMG * W_IMG;

    // Rows/cols 7..15 of the 16x16 WMMA tile must stay zero: init pad once.
    #pragma unroll
    for (int i = 7; i < 16; ++i) xpose[wave][lane][i] = 0.0f;

    v8f acc = {0.0f, 0.0f, 0.0f, 0.0f, 0.0f, 0.0f, 0.0f, 0.0f};

    const int nChunks = (NPIX + 31) / 32;
    const int wid     = gx * 8 + wave;
    const int wstride = gpb * 8;

    const int kSel = lane >> 4;     // 0: K=0/1 half, 1: K=2/3 half (A layout)
    const int row  = lane & 15;     // M (and N) index 0..15

    const float c1f = 2047.0f / 8418.0f;
    const float c2f =  913.0f / 8418.0f;
    const float c3f =  112.0f / 8418.0f;

    for (int t = wid; t < nChunks; t += wstride) {
        const int p = t * 32 + lane;
        float l0 = 0.f, l1 = 0.f, l2 = 0.f, l3 = 0.f, l4 = 0.f, l5 = 0.f, l6 = 0.f;
        if (p < NPIX) {
            const int y = p / CW;
            const int x = p - y * CW;
            const int r = y + BORDER;
            const int c = x + BORDER;
            const float* rp = pIds + (size_t)r * W_IMG + c;
            // 7-tap antisymmetric derivative filter (cross-correlation, like XLA conv)
            float Ix = c1f * (rp[1] - rp[-1]) + c2f * (rp[2] - rp[-2]) + c3f * (rp[3] - rp[-3]);
            float Iy = c1f * (rp[ W_IMG]  - rp[-W_IMG])
                     + c2f * (rp[2*W_IMG] - rp[-2*W_IMG])
                     + c3f * (rp[3*W_IMG] - rp[-3*W_IMG]);
            float Ixv = Ix * PXc;
            float Iyv = Iy * PYc;
            float xs  = ((float)c - U0c) / PXc;
            float ys  = ((float)r - V0c) / PYc;
            float xy  = xs * ys;
            float Zv  = pZ [(size_t)r * W_IMG + c];
            float e   = pIs[(size_t)r * W_IMG + c] - rp[0];
            l0 = Ixv * Zv;
            l1 = Iyv * Zv;
            l2 = -(xs * Ixv + ys * Iyv) * Zv;
            l3 = -Ixv * xy - (1.0f + ys * ys) * Iyv;
            l4 = (1.0f + xs * xs) * Ixv + Iyv * xy;
            l5 = Iyv * xs - Ixv * ys;
            l6 = e;
        }
        float* slot = &xpose[wave][lane][0];
        slot[0] = l0; slot[1] = l1; slot[2] = l2; slot[3] = l3;
        slot[4] = l4; slot[5] = l5; slot[6] = l6;

        // LDS RAW within a single wave is in-order (DScnt); compiler inserts waits.
        // A-matrix f32 16x4 layout: lanes 0-15 M=0..15 {v0:K=0, v1:K=1},
        // lanes 16-31 {v0:K=2, v1:K=3}.  B (4x16) mirrors it for a Gram product,
        // so SRC1 reuses the same registers as SRC0.
        #pragma unroll
        for (int w = 0; w < 8; ++w) {
            const int s0 = 4 * w + 2 * kSel;
            v2f a;
            a.x = xpose[wave][s0    ][row];
            a.y = xpose[wave][s0 + 1][row];
            acc = __builtin_amdgcn_wmma_f32_16x16x4_f32(
                false, a, false, a, (short)0, acc, false, false);
        }
    }

    // C/D layout: VGPR i, lanes 0-15 hold row M=i, N=lane. Rows/cols >=7 are zero.
    if (lane < 7) {
        #pragma unroll
        for (int i = 0; i < 7; ++i) part[wave][i * 7 + lane] = acc[i];
    }
    __syncthreads();

    for (int k = tid; k < 49; k += 256) {
        float s = 0.0f;
        #pragma unroll
        for (int w = 0; w < 8; ++w) s += part[w][k];
        gpart[((size_t)b * gpb + gx) * 49 + k] = s;
    }
}

// Per-batch: sum partials, build H + mu*diag(H), solve 6x6, write -x.
__global__ __launch_bounds__(64)
void dvs_solve_kernel(const float* __restrict__ gpart,
                      const float* __restrict__ mu,
                      float* __restrict__ out, int gpb)
{
    const int b = blockIdx.x * blockDim.x + threadIdx.x;
    if (b >= BATCH) return;

    float S[49];
    #pragma unroll
    for (int k = 0; k < 49; ++k) S[k] = 0.0f;
    for (int g = 0; g < gpb; ++g) {
        const float* p = gpart + ((size_t)b * gpb + g) * 49;
        #pragma unroll
        for (int k = 0; k < 49; ++k) S[k] += p[k];
    }

    float M[6][6], rhs[6];
    const float m = mu[b];
    #pragma unroll
    for (int i = 0; i < 6; ++i) {
        #pragma unroll
        for (int j = 0; j < 6; ++j) M[i][j] = S[i * 7 + j];
        M[i][i] += m * S[i * 7 + i];
        rhs[i] = S[i * 7 + 6];          // g = L^T e lives in augmented column 6
    }

    // Gaussian elimination with partial pivoting (6x6).
    for (int col = 0; col < 6; ++col) {
        int piv = col;
        float best = fabsf(M[col][col]);
        for (int r = col + 1; r < 6; ++r) {
            float v = fabsf(M[r][col]);
            if (v > best) { best = v; piv = r; }
        }
        if (piv != col) {
            for (int j = 0; j < 6; ++j) {
                float tswap = M[col][j]; M[col][j] = M[piv][j]; M[piv][j] = tswap;
            }
            float tswap = rhs[col]; rhs[col] = rhs[piv]; rhs[piv] = tswap;
        }
        const float inv = 1.0f / M[col][col];
        for (int r = col + 1; r < 6; ++r) {
            const float f = M[r][col] * inv;
            for (int j = col; j < 6; ++j) M[r][j] -= f * M[col][j];
            rhs[r] -= f * rhs[col];
        }
    }
    float x[6];
    for (int i = 5; i >= 0; --i) {
        float s = rhs[i];
        for (int j = i + 1; j < 6; ++j) s -= M[i][j] * x[j];
        x[i] = s / M[i][i];
    }
    #pragma unroll
    for (int i = 0; i < 6; ++i) out[b * 6 + i] = -x[i];
}

extern "C" void kernel_launch(void* const* d_in, const int* in_sizes, int n_in,
                              void* d_out, int out_size, void* d_ws, size_t ws_size,
                              hipStream_t stream) {
    const float* Is   = (const float*)d_in[0];
    const float* Ids  = (const float*)d_in[1];
    const float* Zinv = (const float*)d_in[2];
    const float* mu   = (const float*)d_in[3];
    float* out   = (float*)d_out;
    float* gpart = (float*)d_ws;

    int gpb = 8;                                  // groups per batch
    const size_t slot = (size_t)BATCH * 49 * sizeof(float);
    if (ws_size < (size_t)gpb * slot) {
        gpb = (int)(ws_size / slot);
        if (gpb < 1) gpb = 1;
    }

    dim3 grid(gpb, BATCH);
    dvs_gram_kernel<<<grid, 256, 0, stream>>>(Is, Ids, Zinv, gpart, gpb);
    dvs_solve_kernel<<<1, 64, 0, stream>>>(gpart, mu, out, gpb);
    (void)in_sizes; (void)n_in; (void)out_size;
}